// MF_17386027614868
// MI455X (gfx1250) — compile-verified
//
#include <hip/hip_runtime.h>

// CDNA5 / gfx1250, wave32. Matrix-factorization inference:
//   pred[b] = dot(uw[user[b]] + ubias[user[b]], iw[item[b]] + ibias[item[b]]) + bias0
//
// Memory-bound (~558 MB @ 23.3 TB/s => ~24 us floor; 136 MFLOP is noise).
// Strategy: 16 pairs per wave; per K-chunk of 4, each lane loads one float2 of
// the user row and one float2 of the item row (A-matrix 16x4 f32 WMMA layout),
// forms the elementwise products, and reduces across the hidden dim with
// V_WMMA_F32_16X16X4_F32 against an all-ones B matrix (row-sum trick).

typedef float v2f __attribute__((ext_vector_type(2)));
typedef float v4f __attribute__((ext_vector_type(4)));
typedef float v8f __attribute__((ext_vector_type(8)));

#define HIDDEN 64
#define PAIRS_PER_WAVE 16
#define WAVES_PER_BLOCK 8

__global__ __launch_bounds__(256) void mf_dot_wmma_kernel(
    const int* __restrict__ user,
    const int* __restrict__ item,
    const float* __restrict__ user_weight,
    const float* __restrict__ item_weight,
    const float* __restrict__ user_bias,
    const float* __restrict__ item_bias,
    const float* __restrict__ bias,
    float* __restrict__ out,
    int batch)
{
    const int lane = (int)(threadIdx.x & 31u);   // wave32
    const int wave = (int)(threadIdx.x >> 5);    // 0..7
    const int pl   = lane & 15;                  // pair-in-tile == M row of A
    const int hi   = lane >> 4;                  // which K-half this lane feeds

    // Tile of 16 consecutive pairs handled by this wave.
    const long tile = ((long)blockIdx.x * WAVES_PER_BLOCK + wave) * PAIRS_PER_WAVE;
    long pidx = tile + pl;
    // Clamp without branching (EXEC must stay all-ones for WMMA).
    pidx = (pidx < (long)batch) ? pidx : (long)(batch - 1);

    const int u  = user[pidx];
    const int it = item[pidx];
    const float bu = user_bias[u];
    const float bi = item_bias[it];

    const float* __restrict__ urow = user_weight + (size_t)u  * HIDDEN;
    const float* __restrict__ irow = item_weight + (size_t)it * HIDDEN;

    v8f acc = {};                 // D[m][n] accumulates row-sums (identical over n)
    const v2f ones = {1.0f, 1.0f};  // B = all-ones 4x16 -> row-sum reduction

    // K=64 hidden dim as 16 chunks of K=4. A-matrix f32 16x4 layout:
    // VGPR0 = {K0 lanes0-15 | K2 lanes16-31}, VGPR1 = {K1 | K3}, so lane (pl,hi)
    // supplies the consecutive element pair at k = 4j + 2*hi -> one b64 load
    // per matrix per chunk; the two lane-halves jointly consume each 16B of row.
    #pragma unroll
    for (int j = 0; j < 16; ++j) {
        const int k = j * 4 + hi * 2;
        const v2f ue = *(const v2f*)(urow + k);
        const v2f ie = *(const v2f*)(irow + k);
        v2f a;
        a.x = (ue.x + bu) * (ie.x + bi);
        a.y = (ue.y + bu) * (ie.y + bi);
        acc = __builtin_amdgcn_wmma_f32_16x16x4_f32(
            /*neg_a=*/false, a, /*neg_b=*/false, ones,
            /*c_mod=*/(short)0, acc, /*reuse_a=*/false, /*reuse_b=*/false);
    }

    const float b0 = bias[0];
    #pragma unroll
    for (int r = 0; r < 8; ++r) acc[r] += b0;

    // D layout: lane 0 holds pred[tile+0..7] in its 8 acc VGPRs,
    // lane 16 holds pred[tile+8..15]. Two b128 stores per contributing lane.
    if (pl == 0) {
        const long ob = tile + (long)hi * 8;
        if (ob + 8 <= (long)batch) {
            v4f lo = {acc[0], acc[1], acc[2], acc[3]};
            v4f hi4 = {acc[4], acc[5], acc[6], acc[7]};
            *(v4f*)(out + ob)     = lo;
            *(v4f*)(out + ob + 4) = hi4;
        } else {
            #pragma unroll
            for (int r = 0; r < 8; ++r)
                if (ob + r < (long)batch) out[ob + r] = acc[r];
        }
    }
}

extern "C" void kernel_launch(void* const* d_in, const int* in_sizes, int n_in,
                              void* d_out, int out_size, void* d_ws, size_t ws_size,
                              hipStream_t stream) {
    const int*   user = (const int*)d_in[0];
    const int*   item = (const int*)d_in[1];
    const float* uw   = (const float*)d_in[2];
    const float* iw   = (const float*)d_in[3];
    const float* ub   = (const float*)d_in[4];
    const float* ib   = (const float*)d_in[5];
    const float* bias = (const float*)d_in[6];
    float* out = (float*)d_out;

    const int batch  = in_sizes[0];
    const int waves  = (batch + PAIRS_PER_WAVE - 1) / PAIRS_PER_WAVE;
    const int blocks = (waves + WAVES_PER_BLOCK - 1) / WAVES_PER_BLOCK;

    mf_dot_wmma_kernel<<<blocks, 256, 0, stream>>>(
        user, item, uw, iw, ub, ib, bias, out, batch);
}